// OmniDIBR_84713934946657
// MI455X (gfx1250) — compile-verified
//
#include <hip/hip_runtime.h>
#include <stdint.h>

// Problem geometry (fixed by setup_inputs)
#define BDIM 8
#define HDIM 512
#define WDIM 1024
#define NPIX (HDIM * WDIM)        // 524288
#define TOTAL_PIX (BDIM * NPIX)   // 4194304
#define XI_C 0.9f
#define MIND 1.0f
#define MAXD 100.0f

// Work decomposition: 8 waves/block, each wave does 4 tiles of 128 px (one half-row)
#define TILE_PX 128
#define TPW 4
#define WPB 8
#define PIX_PER_WAVE (TILE_PX * TPW)                    // 512
#define NBLOCKS (TOTAL_PIX / (WPB * PIX_PER_WAVE))      // 1024
#define NBUF 3                                          // 3-deep async pipeline
#define BUFB (3 * 512)                                  // bytes per tile buffer (3 streams)

// CDNA5 async global->LDS path (ASYNCcnt), guarded so host pass still compiles.
#if defined(__AMDGCN__) && __has_builtin(__builtin_amdgcn_global_load_async_to_lds_b128) && __has_builtin(__builtin_amdgcn_s_wait_asynccnt)
#define USE_ASYNC 1
#else
#define USE_ASYNC 0
#endif

#if USE_ASYNC
typedef int v4i_t __attribute__((ext_vector_type(4)));
typedef __attribute__((address_space(1))) v4i_t* g_v4i_p;   // global int4*
typedef __attribute__((address_space(3))) v4i_t* l_v4i_p;   // LDS int4*
__device__ __forceinline__ void async_cp16(void* lds_dst, const void* gsrc) {
  // (global src, lds dst, imm offset, imm cpol)
  __builtin_amdgcn_global_load_async_to_lds_b128((g_v4i_p)gsrc, (l_v4i_p)lds_dst, 0, 0);
}
#endif

__global__ void __launch_bounds__(WPB * 32)
omni_dibr_kernel(const float* __restrict__ depth,
                 const float* __restrict__ invK,
                 const float* __restrict__ Kmat,
                 const float* __restrict__ ncoef,
                 const int* __restrict__ nsa,
                 const float* __restrict__ Tmat,
                 float* __restrict__ out)
{
#if USE_ASYNC
  // per wave: 3 buffers x 3 streams x 512B = 4.5KB -> 36KB/block (of 320KB/WGP)
  __shared__ __align__(16) unsigned char lds_buf[WPB * NBUF * BUFB];
#endif
  const int lane = (int)(threadIdx.x & 31u);
  const int wib  = (int)(threadIdx.x >> 5);
  const int gw   = (int)blockIdx.x * WPB + wib;
  const int wavebase = gw * PIX_PER_WAVE;     // global pixel id of wave start
  int b = wavebase / NPIX;                    // batch, wave-uniform
#if defined(__AMDGCN__)
  b = __builtin_amdgcn_readfirstlane(b);      // scalarize matrix loads (s_load nv)
#endif
  const int nb0 = wavebase - b * NPIX;        // pixel-in-batch base (half-row aligned)
  const int y = nb0 >> 10;                    // W == 1024
  const float py = (float)y;

  // Wave-uniform 3x3 matrices (tiled per batch in the inputs)
  const float* ikp = invK + b * 16;
  const float ik00 = ikp[0], ik01 = ikp[1], ik02 = ikp[2];
  const float ik10 = ikp[4], ik11 = ikp[5], ik12 = ikp[6];
  const float ik20 = ikp[8], ik21 = ikp[9], ik22 = ikp[10];
  const float* kp = Kmat + b * 16;
  const float k00 = kp[0], k01 = kp[1], k02 = kp[2];
  const float k10 = kp[4], k11 = kp[5], k12 = kp[6];
  const float* tp = Tmat + b * 16;
  const float r00 = tp[0], r01 = tp[1], r02 = tp[2], t0 = tp[3];
  const float r10 = tp[4], r11 = tp[5], r12 = tp[6], t1 = tp[7];
  const float r20 = tp[8], r21 = tp[9], r22 = tp[10], t2 = tp[11];

  const float* dpb = depth + b * NPIX;
  const float* ncb = ncoef + b * NPIX;
  const int*   sab = nsa;   // masks use batch-0 plane only (reference: reshape(B,-1)[0])
  float* pccb = out + (size_t)(b * HDIM + y) * (size_t)(WDIM * 2);
  float* cecb = out + (size_t)TOTAL_PIX * 2 + (size_t)b * 3 * NPIX;

#if USE_ASYNC
  unsigned char* myl = lds_buf + wib * (NBUF * BUFB);
#define ISSUE_TILE(t_)                                                  \
  {                                                                     \
    const int nb_ = nb0 + (t_) * TILE_PX;                               \
    unsigned char* L_ = myl + ((t_) % NBUF) * BUFB + lane * 16;         \
    async_cp16(L_,        dpb + nb_ + lane * 4);                        \
    async_cp16(L_ + 512,  ncb + nb_ + lane * 4);                        \
    async_cp16(L_ + 1024, sab + nb_ + lane * 4);                        \
  }
#endif

  auto compute_tile = [&](int t) {
    const int npx = nb0 + t * TILE_PX + lane * 4;  // first of this lane's 4 pixels
    const int xb  = npx & (WDIM - 1);
    float4 d4, nc4;
    int4 s4;
#if USE_ASYNC
    {
      const unsigned char* L = myl + (t % NBUF) * BUFB + lane * 16;
      d4  = *(const float4*)(L);
      nc4 = *(const float4*)(L + 512);
      s4  = *(const int4*)(L + 1024);
    }
#else
    d4  = *(const float4*)(dpb + npx);
    nc4 = *(const float4*)(ncb + npx);
    s4  = *(const int4*)(sab + npx);
#if defined(__AMDGCN__)
    if (t + 1 < TPW) {  // lowers to global_prefetch_b8 on gfx1250
      __builtin_prefetch(dpb + npx + TILE_PX, 0, 1);
      __builtin_prefetch(ncb + npx + TILE_PX, 0, 1);
      __builtin_prefetch(sab + npx + TILE_PX, 0, 1);
    }
#endif
#endif
    float opcc[8] __attribute__((aligned(16)));
    float cc0[4]  __attribute__((aligned(16)));
    float cc1[4]  __attribute__((aligned(16)));
    float cc2[4]  __attribute__((aligned(16)));
#pragma unroll
    for (int jj = 0; jj < 4; ++jj) {
      const float dd = ((const float*)&d4)[jj];
      const float nc = ((const float*)&nc4)[jj];
      const int   ss = ((const int*)&s4)[jj];
      const float px = (float)(xb + jj);
      // cam = inv_K[:3,:3] @ (px, py, 1)
      const float cam0 = fmaf(ik00, px, fmaf(ik01, py, ik02));
      const float cam1 = fmaf(ik10, px, fmaf(ik11, py, ik12));
      const float cam2 = fmaf(ik20, px, fmaf(ik21, py, ik22));
      const float m0 = nc * cam0, m1 = nc * cam1, m2 = nc * cam2;  // nc*cam
      const float dm1 = dd - 1.0f;
      // x_s1 = d*nc*cam - (d-1)*t_xi ; t_xi = (0,0,XI)
      const float xs10 = dd * m0;
      const float xs11 = dd * m1;
      const float xs12 = fmaf(dd, m2, -(dm1 * XI_C));
      const float x12  = xs12 - XI_C;                 // x_1 z-component
      // Tx = R @ x_s1 + t
      const float Tx0 = fmaf(r00, xs10, fmaf(r01, xs11, fmaf(r02, xs12, t0)));
      const float Tx1 = fmaf(r10, xs10, fmaf(r11, xs11, fmaf(r12, xs12, t1)));
      const float Tx2 = fmaf(r20, xs10, fmaf(r21, xs11, fmaf(r22, xs12, t2)));
      const float x22 = Tx2 - XI_C;                   // x_2 z-component
      const float n1 = sqrtf(fmaf(xs10, xs10, fmaf(xs11, xs11, x12 * x12)));
      const float n2 = sqrtf(fmaf(Tx0, Tx0, fmaf(Tx1, Tx1, x22 * x22)));
      const float a1 = (n1 - MIND) / (MAXD - MIND);
      const float a2 = (n2 - MIND) / (MAXD - MIND);
      // item2_min = (MIN_DEPTH-1)*t_xi = 0, so x_min = m, x_max = 100*m
      const float s2m0 = fmaf(r00, m0, fmaf(r01, m1, fmaf(r02, m2, t0)));
      const float s2m1 = fmaf(r10, m0, fmaf(r11, m1, fmaf(r12, m2, t1)));
      const float s2m2 = fmaf(r20, m0, fmaf(r21, m1, fmaf(r22, m2, t2)));
      const float M0 = MAXD * m0, M1 = MAXD * m1, M2 = MAXD * m2;
      const float s2M0 = fmaf(r00, M0, fmaf(r01, M1, fmaf(r02, M2, t0)));
      const float s2M1 = fmaf(r10, M0, fmaf(r11, M1, fmaf(r12, M2, t1)));
      const float s2M2 = fmaf(r20, M0, fmaf(r21, M1, fmaf(r22, M2, t2)));
      // x_2_dash = Tx + (n2-1)*t_xi ; p2 = (K @ x_2_dash) / z
      const float x2d2 = fmaf(n2 - 1.0f, XI_C, Tx2);
      float p2x = fmaf(k00, Tx0, fmaf(k01, Tx1, k02 * x2d2)) / x2d2;
      float p2y = fmaf(k10, Tx0, fmaf(k11, Tx1, k12 * x2d2)) / x2d2;
      const bool outside = (ss == 0) || (ss == -1);
      const bool inside  = (ss == 1);
      if (outside) { p2x = px; p2y = py; }
      opcc[jj * 2]     = (p2x / (float)(WDIM - 1) - 0.5f) * 2.0f;
      opcc[jj * 2 + 1] = (p2y / (float)(HDIM - 1) - 0.5f) * 2.0f;
      // idx = clip(int(round(p2x) + round(p2y)*H), 0, N-1)  (note: H, per reference)
      const float qf = rintf(p2x) + rintf(p2y) * (float)HDIM;
      int gidx = __float2int_rz(qf);
      gidx = gidx < 0 ? 0 : (gidx > NPIX - 1 ? NPIX - 1 : gidx);
      const float gx = (float)(gidx & (WDIM - 1));
      const float gy = (float)(gidx >> 10);
      const float ncg = ncb[gidx];                    // data-dependent gather
      const float cg0 = fmaf(ik00, gx, fmaf(ik01, gy, ik02));
      const float cg1 = fmaf(ik10, gx, fmaf(ik11, gy, ik12));
      const float cg2 = fmaf(ik20, gx, fmaf(ik21, gy, ik22));
      const float g0 = ncg * cg0, g1 = ncg * cg1, g2 = ncg * cg2; // gmin; gmax=100*g
      const float b1 = 1.0f - a1, b2 = 1.0f - a2;
      float c0v = fmaf(b1, s2m0, a1 * s2M0) - fmaf(b2, g0, a2 * (MAXD * g0));
      float c1v = fmaf(b1, s2m1, a1 * s2M1) - fmaf(b2, g1, a2 * (MAXD * g1));
      float c2v = fmaf(b1, s2m2, a1 * s2M2) - fmaf(b2, g2, a2 * (MAXD * g2));
      if (!inside) { c0v = 0.0f; c1v = 0.0f; c2v = 0.0f; }
      cc0[jj] = c0v; cc1[jj] = c1v; cc2[jj] = c2v;
    }
    // Coalesced 16B stores for both outputs
    *(float4*)(pccb + (size_t)xb * 2)     = *(const float4*)(&opcc[0]);
    *(float4*)(pccb + (size_t)xb * 2 + 4) = *(const float4*)(&opcc[4]);
    *(float4*)(cecb + npx)                = *(const float4*)(&cc0[0]);
    *(float4*)(cecb + NPIX + npx)         = *(const float4*)(&cc1[0]);
    *(float4*)(cecb + 2 * NPIX + npx)     = *(const float4*)(&cc2[0]);
  };

#if USE_ASYNC
  // 3-deep software pipeline: copies for tile t are issued two tiles ahead of
  // consumption. s_wait_asynccnt requires literal immediates, so the pipeline
  // is laid out so every wait is a compile-time constant. Async loads complete
  // in issue order, so asynccnt<=6 means the oldest tile's 3 copies landed.
  ISSUE_TILE(0)
  ISSUE_TILE(1)
#pragma unroll
  for (int t = 0; t < TPW - 2; ++t) {
    ISSUE_TILE(t + 2)
    __builtin_amdgcn_s_wait_asynccnt(6);   // tile t's 3 copies complete
    compute_tile(t);
  }
  __builtin_amdgcn_s_wait_asynccnt(3);     // tile TPW-2 complete
  compute_tile(TPW - 2);
  __builtin_amdgcn_s_wait_asynccnt(0);     // tile TPW-1 complete
  compute_tile(TPW - 1);
#else
#pragma unroll
  for (int t = 0; t < TPW; ++t) compute_tile(t);
#endif
}

extern "C" void kernel_launch(void* const* d_in, const int* in_sizes, int n_in,
                              void* d_out, int out_size, void* d_ws, size_t ws_size,
                              hipStream_t stream) {
  (void)in_sizes; (void)n_in; (void)out_size; (void)d_ws; (void)ws_size;
  const float* depth = (const float*)d_in[0];
  const float* invK  = (const float*)d_in[1];
  const float* Kmat  = (const float*)d_in[2];
  const float* ncoef = (const float*)d_in[3];
  const int*   nsa   = (const int*)d_in[4];
  const float* Tmat  = (const float*)d_in[5];
  float* out = (float*)d_out;
  hipLaunchKernelGGL(omni_dibr_kernel, dim3(NBLOCKS), dim3(WPB * 32), 0, stream,
                     depth, invK, Kmat, ncoef, nsa, Tmat, out);
}